// ChamferDistanceL2_52115133170347
// MI455X (gfx1250) — compile-verified
//
#include <hip/hip_runtime.h>
#include <math.h>

// Problem constants (from reference): B=64, S=128, V=4096, PAD=4096, EOS=0
#define NB 64
#define NS 128
#define NV 4096
#define PADTOK 4096

typedef __attribute__((ext_vector_type(2))) float v2f;
typedef __attribute__((ext_vector_type(8))) float v8f;
typedef __attribute__((ext_vector_type(4))) int v4i;

// ---------------- CDNA5 async global->LDS staging ----------------
#if defined(__has_builtin)
#if __has_builtin(__builtin_amdgcn_global_load_async_to_lds_b128)
#define HAVE_ASYNC_LDS 1
#endif
#if __has_builtin(__builtin_amdgcn_s_wait_asynccnt)
#define HAVE_WAIT_ASYNC 1
#endif
#endif

#ifdef HAVE_ASYNC_LDS
typedef __attribute__((address_space(1))) v4i as1_v4i;
typedef __attribute__((address_space(3))) v4i as3_v4i;
#endif

__device__ __forceinline__ void async_load_16B(const float* gsrc, float* ldst) {
#ifdef HAVE_ASYNC_LDS
  __builtin_amdgcn_global_load_async_to_lds_b128(
      (as1_v4i*)(void*)gsrc, (as3_v4i*)(void*)ldst, /*offset=*/0, /*cpol=*/0);
#else
  *(float4*)ldst = *(const float4*)gsrc;  // synchronous fallback
#endif
}

__device__ __forceinline__ void wait_async_lds() {
#ifdef HAVE_ASYNC_LDS
#ifdef HAVE_WAIT_ASYNC
  __builtin_amdgcn_s_wait_asynccnt(0);
#else
  asm volatile("s_wait_asynccnt 0x0" ::: "memory");
#endif
#endif
}

// ---------------- block reductions (256 threads) ----------------
__device__ __forceinline__ float block_sum_256(float v, float* red) {
  const int tid = threadIdx.x;
  red[tid] = v;
  __syncthreads();
  for (int off = 128; off > 0; off >>= 1) {
    if (tid < off) red[tid] += red[tid + off];
    __syncthreads();
  }
  float r = red[0];
  __syncthreads();
  return r;
}

__device__ __forceinline__ float block_max_256(float v, float* red) {
  const int tid = threadIdx.x;
  red[tid] = v;
  __syncthreads();
  for (int off = 128; off > 0; off >>= 1) {
    if (tid < off) red[tid] = fmaxf(red[tid], red[tid + off]);
    __syncthreads();
  }
  float r = red[0];
  __syncthreads();
  return r;
}

// ===================================================================
// Kernel 1: per-row softmax statistics + one-hot gather.
// One block per (b, s) row of 4096 logits. Row is staged in LDS via
// async copies. Emits: eos_prob, SS = sum_{c>=1} p_c^2, and the 128
// gathered probabilities g[row][j] = p_row[targets[b, j]].
// ===================================================================
__global__ __launch_bounds__(256) void softmax_stats_kernel(
    const float* __restrict__ logits, const int* __restrict__ targets,
    float* __restrict__ eosp, float* __restrict__ ss, float* __restrict__ gmat) {
  __shared__ __align__(16) float srow[NV];
  __shared__ float red[256];

  const int row = blockIdx.x;  // 0 .. B*S-1
  const int b = row >> 7;
  const int tid = threadIdx.x;
  const float* src = logits + (size_t)row * NV;

  // Stage 16KB row into LDS: 4 x float4 per thread, coalesced.
  for (int it = 0; it < 4; ++it) {
    const int idx = it * 1024 + tid * 4;
    async_load_16B(src + idx, &srow[idx]);
  }
  wait_async_lds();
  __syncthreads();

  // 1) row max
  float m = -__builtin_inff();
  for (int it = 0; it < 16; ++it) m = fmaxf(m, srow[tid + it * 256]);
  m = block_max_256(m, red);

  // 2) Z = sum exp, SSraw = sum_{c>=1} exp^2
  float sum = 0.f, sq = 0.f;
  for (int it = 0; it < 16; ++it) {
    const int c = tid + it * 256;
    const float e = __expf(srow[c] - m);
    sum += e;
    sq += (c == 0) ? 0.f : e * e;
  }
  sum = block_sum_256(sum, red);
  sq = block_sum_256(sq, red);
  const float invZ = 1.0f / sum;

  if (tid == 0) {
    eosp[row] = __expf(srow[0] - m) * invZ;  // p(class 0), pre-mask
    ss[row] = sq * invZ * invZ;              // sum_{c>=1} p_c^2
  }
  // 3) gather p[targets[b, j]] for the 128 positions of this batch
  if (tid < NS) {
    const int t = targets[b * NS + tid];
    float gv = 0.f;
    if (t >= 1 && t <= NV - 1) gv = __expf(srow[t] - m) * invZ;
    gmat[(size_t)row * NS + tid] = gv;
  }
}

// ===================================================================
// Kernel 2: one block per batch. Builds d2[i][j] tiles with
// v_wmma_f32_16x16x4_f32:  A=[x2_i | 1], B=[1 ; y2_j], C=-2*G tile,
// so D = x2_i + y2_j - 2*g_ij exactly (f32 WMMA). Then min-reductions
// for the chamfer terms, plus the EOS BCE sums.
// ===================================================================
__global__ __launch_bounds__(256) void chamfer_eos_kernel(
    const int* __restrict__ targets, const float* __restrict__ eosp,
    const float* __restrict__ ss, const float* __restrict__ gmat,
    float* __restrict__ lbl, float* __restrict__ eosl) {
  __shared__ float x2s[NS];
  __shared__ float y2s[NS];
  __shared__ float rowmin[NS];
  __shared__ float colmin[NS];
  __shared__ float tile[NS][36];  // 32 active cols + pad
  __shared__ float red[256];

  const int b = blockIdx.x;
  const int tid = threadIdx.x;
  const int lane = tid & 31;
  const int wave = tid >> 5;
  const int half = lane >> 4;  // 0: lanes 0-15, 1: lanes 16-31
  const int l = lane & 15;

  if (tid < NS) {
    const int t = targets[b * NS + tid];
    const float mask = (t != 0 && t != PADTOK) ? 1.f : 0.f;  // eos_head
    y2s[tid] = mask;                                         // |one_hot*mask|^2
    x2s[tid] = mask * ss[b * NS + tid];                      // |probs_m[:,1:]|^2
    rowmin[tid] = __builtin_inff();
  }
  __syncthreads();

  const float* gb = gmat + ((size_t)b << 14);  // b*128*128

  for (int cg = 0; cg < 4; ++cg) {  // 4 column-groups of 32 j's
    for (int k = 0; k < 2; ++k) {   // wave -> tile-row, k -> tile-col in group
      const int i0 = wave * 16;
      const int j0 = cg * 32 + k * 16;

      // A (16x4 f32): K0 = x2_i, K1 = 1, K2 = K3 = 0
      v2f a;
      a.x = (half == 0) ? x2s[i0 + l] : 0.f;  // lanes 0-15: K=0 ; 16-31: K=2
      a.y = (half == 0) ? 1.0f : 0.f;         // lanes 0-15: K=1 ; 16-31: K=3
      // B (4x16 f32): row K0 = 1, row K1 = y2_j, rows K2,K3 = 0
      v2f bb;
      bb.x = (half == 0) ? 1.0f : 0.f;
      bb.y = (half == 0) ? y2s[j0 + l] : 0.f;
      // C = -2 * G tile in C/D layout (vgpr r: M = r + 8*half, N = l)
      v8f c;
      for (int r = 0; r < 8; ++r) {
        const int i = i0 + r + half * 8;
        c[r] = -2.0f * gb[i * NS + (j0 + l)];
      }
      v8f d = __builtin_amdgcn_wmma_f32_16x16x4_f32(
          false, a, false, bb, (short)0, c, false, false);
      for (int r = 0; r < 8; ++r) {
        const int i = i0 + r + half * 8;
        tile[i][(j0 + l) - cg * 32] = d[r];
      }
    }
    __syncthreads();
    if (tid < NS) {  // update running row mins (min over j)
      float rm = rowmin[tid];
      for (int jj = 0; jj < 32; ++jj) rm = fminf(rm, tile[tid][jj]);
      rowmin[tid] = rm;
    } else if (tid < NS + 32) {  // column mins (min over i), complete per group
      const int jl = tid - NS;
      float cm = __builtin_inff();
      for (int ii = 0; ii < NS; ++ii) cm = fminf(cm, tile[ii][jl]);
      colmin[cg * 32 + jl] = cm;
    }
    __syncthreads();
  }

  // label part: (mean_i rowmin + mean_j colmin)
  float v = (tid < NS) ? (rowmin[tid] + colmin[tid]) : 0.f;
  const float lsum = block_sum_256(v, red);
  if (tid == 0) lbl[b] = lsum / (float)NS;

  // EOS BCE part
  float bce = 0.f, epf = 0.f, ehf = 0.f;
  if (tid < NS) {
    const int t = targets[b * NS + tid];
    const float p = eosp[b * NS + tid];
    const float tt = (t == 0 || t == PADTOK) ? 1.f : 0.f;  // eos_target
    float lp = fmaxf(logf(p), -100.f);
    float l1 = fmaxf(log1pf(-p), -100.f);
    bce = -(tt * lp + (1.f - tt) * l1);
    epf = (t == 0) ? 1.f : 0.f;  // eos_pos
    ehf = 1.f - tt;              // eos_head
  }
  const float s_be = block_sum_256(bce * epf, red);
  const float s_ep = block_sum_256(epf, red);
  const float s_bh = block_sum_256(bce * ehf, red);
  const float s_eh = block_sum_256(ehf, red);
  if (tid == 0)
    eosl[b] = 0.5f * s_be / (s_ep + 1e-8f) + 0.5f * s_bh / (s_eh + 1e-8f);
}

// ===================================================================
// Kernel 3: means over 64 batches -> out[0..1]
// ===================================================================
__global__ __launch_bounds__(64) void finalize_kernel(
    const float* __restrict__ lbl, const float* __restrict__ eosl,
    float* __restrict__ out) {
  __shared__ float r1[64];
  __shared__ float r2[64];
  const int tid = threadIdx.x;
  r1[tid] = lbl[tid];
  r2[tid] = eosl[tid];
  __syncthreads();
  for (int off = 32; off > 0; off >>= 1) {
    if (tid < off) {
      r1[tid] += r1[tid + off];
      r2[tid] += r2[tid + off];
    }
    __syncthreads();
  }
  if (tid == 0) {
    out[0] = r1[0] / (float)NB;  // label_loss
    out[1] = r2[0] / (float)NB;  // eos_loss
  }
}

extern "C" void kernel_launch(void* const* d_in, const int* in_sizes, int n_in,
                              void* d_out, int out_size, void* d_ws, size_t ws_size,
                              hipStream_t stream) {
  const float* logits = (const float*)d_in[0];
  const int* targets = (const int*)d_in[1];

  // Workspace layout (floats): [eosp: 8192][ss: 8192][gmat: 8192*128][lbl: 64][eosl: 64]
  float* ws = (float*)d_ws;
  float* eosp = ws;
  float* ss = ws + (NB * NS);
  float* gmat = ws + 2 * (NB * NS);
  float* lbl = gmat + (size_t)NB * NS * NS;
  float* eosl = lbl + NB;

  softmax_stats_kernel<<<NB * NS, 256, 0, stream>>>(logits, targets, eosp, ss, gmat);
  chamfer_eos_kernel<<<NB, 256, 0, stream>>>(targets, eosp, ss, gmat, lbl, eosl);
  finalize_kernel<<<1, 64, 0, stream>>>(lbl, eosl, (float*)d_out);
}